// NLSearch_47906065219884
// MI455X (gfx1250) — compile-verified
//
#include <hip/hip_runtime.h>
#include <hip/hip_bf16.h>

// ---------------------------------------------------------------------------
// Non-local patch search (NLSearch) for MI455X / gfx1250.
//
// dist[n,v,s] = (Sh * CC_s * Sw^T)[n,v] in the interior, where
//   CC_s[y,x]  = sum_c vid0[c,y,x] * vid1[c, refl(y+s0'), refl(x+s1')]
//   Sh[n,y]=1 iff |y-4n|<=3,  Sw[v,x]=1 iff |x-4v|<=3
// Both GEMM stages run on V_WMMA_F32_16X16X4_F32; boundary queries
// (n or v in {0,1,63}) are recomputed directly; then stable top-7.
// ---------------------------------------------------------------------------

typedef __attribute__((ext_vector_type(2))) float v2f;
typedef __attribute__((ext_vector_type(8))) float v8f;

#define HW        256          // H == W == 256
#define CCH       32           // channels
#define TT        3            // T
#define NQ        64           // queries per axis (stride 4)
#define NSHIFT    64           // 8x8 shifts
#define NQUERIES  (TT * NQ * NQ)   // 12288
#define KTOP      7
#define T1_STRIDE 260          // 256 + 4 pad (bank-conflict avoidance)

__device__ __forceinline__ int refl(int i) {
    i = (i < 0) ? -i : i;
    return (i >= HW) ? (2 * (HW - 1) - i) : i;   // 510 - i
}

// ---------------------------------------------------------------------------
// Kernel 1: CC[t][s][y][x] = sum_c vid0[t,c,y,x] * vid1[t,c,refl(y+s0'),refl(x+s1')]
// One block = one (y, shift-group, t); thread = x. vid0 pixel held in regs
// and reused across 8 shifts. Whole working set is L2-resident (50 MB << 192 MB).
// ---------------------------------------------------------------------------
__global__ __launch_bounds__(256) void nls_cc(const float* __restrict__ vid0,
                                              const float* __restrict__ vid1,
                                              float* __restrict__ cc) {
    const int x  = threadIdx.x;          // 0..255
    const int y  = blockIdx.x;           // 0..255
    const int sg = blockIdx.y;           // 0..7  (shift group of 8)
    const int t  = blockIdx.z;           // 0..2

    const float* v0 = vid0 + ((size_t)(t * CCH) << 16) + y * HW + x;
    float r0[CCH];
#pragma unroll
    for (int c = 0; c < CCH; ++c) r0[c] = v0[(size_t)c << 16];

    const float* b1 = vid1 + ((size_t)(t * CCH) << 16);
    for (int si = 0; si < 8; ++si) {
        const int s   = sg * 8 + si;
        const int s0p = (s >> 3) - 4;
        const int s1p = (s & 7) - 4;
        const int ry  = refl(y + s0p);
        const int rx  = refl(x + s1p);
        const float* v1 = b1 + ry * HW + rx;
        float acc = 0.0f;
#pragma unroll
        for (int c = 0; c < CCH; ++c) acc += r0[c] * v1[(size_t)c << 16];
        cc[((size_t)(t * NSHIFT + s) << 16) + y * HW + x] = acc;
    }
}

// ---------------------------------------------------------------------------
// Kernel 2: per (t, s): D(64x64) = Sh(64x256) * CC_s(256x256) * Sw^T(256x64)
// via V_WMMA_F32_16X16X4_F32, selection matrices generated in-register,
// K-range pruned to the 68 rows/cols a 16-wide tile can touch.
//
// All loop control is forced wave-uniform via readfirstlane so EXEC stays
// all-ones around the WMMAs (ISA 7.12 requirement) and loop control is SALU.
//
// Fragment layouts (ISA 7.12.2, wave32):
//   A 16x4 : lanes 0-15 -> M=lane, VGPR0=K0, VGPR1=K1; lanes 16-31 -> K2,K3
//   B 4x16 : VGPR0: lanes0-15=(K0,N=lane), lanes16-31=(K2,...); VGPR1: K1/K3
//   C/D    : VGPR r: lanes0-15=(M=r,N=lane), lanes16-31=(M=r+8,...)
// ---------------------------------------------------------------------------
__global__ __launch_bounds__(256) void nls_boxsum_wmma(const float* __restrict__ cc,
                                                       float* __restrict__ dsc) {
    extern __shared__ float T1[];        // 64 rows x T1_STRIDE

    const int bs   = blockIdx.x;         // 0..191
    const int t    = bs >> 6;
    const int s    = bs & 63;
    const float* ccS = cc + ((size_t)(t * NSHIFT + s) << 16);

    const int tid   = threadIdx.x;
    // Wave id as a true scalar: uniform loop bounds -> no EXEC masking.
    const int wave  = __builtin_amdgcn_readfirstlane(tid >> 5);   // 0..7
    const int lane  = tid & 31;
    const int lhalf = lane >> 4;         // 0/1
    const int l16   = lane & 15;

    // ---- Stage 1: T1[n, x] = (Sh * CC)[n, x] ----
    for (int tile = wave; tile < 64; tile += 8) {
        const int nt = tile >> 4;        // 0..3  (16 n-rows)
        const int xt = tile & 15;        // 0..15 (16 x-cols)
        const int x    = xt * 16 + l16;  // B/D column
        const int nrow = nt * 16 + l16;  // A row (query n)
        const int lokn = 4 * nrow - 3;
        const int hikn = 4 * nrow + 3;

        v8f acc = {};
        const int y0 = (nt == 0) ? 0 : (nt * 64 - 4);
        const int y1 = nt * 64 + 64;
#pragma unroll 2
        for (int yb = y0; yb < y1; yb += 4) {
            const int ka = yb + lhalf * 2;      // this lane's K rows: ka, ka+1
            // Prefetch the CC rows two k-steps ahead (global_prefetch_b8;
            // speculative, OOB silently dropped).
            __builtin_prefetch(&ccS[(size_t)(ka + 8) * HW + x], 0, 1);
            v2f a, b;
            a.x = (ka     >= lokn && ka     <= hikn) ? 1.0f : 0.0f;
            a.y = (ka + 1 >= lokn && ka + 1 <= hikn) ? 1.0f : 0.0f;
            b.x = ccS[(size_t)(ka    ) * HW + x];
            b.y = ccS[(size_t)(ka + 1) * HW + x];
            acc = __builtin_amdgcn_wmma_f32_16x16x4_f32(
                false, a, false, b, (short)0, acc, false, false);
        }
#pragma unroll
        for (int r = 0; r < 8; ++r) {
            const int row = nt * 16 + r + lhalf * 8;
            T1[row * T1_STRIDE + x] = acc[r];
        }
    }

    __syncthreads();

    // ---- Stage 2: D[n, v] = sum_x T1[n,x] * Sw[v,x] ----
    for (int tile = wave; tile < 16; tile += 8) {
        const int nt = tile >> 2;        // 0..3
        const int vt = tile & 3;         // 0..3
        const int vcol = vt * 16 + l16;  // B/D column (query v)
        const int nrow = nt * 16 + l16;  // A row (query n)
        const int lokv = 4 * vcol - 3;
        const int hikv = 4 * vcol + 3;

        v8f acc = {};
        const int x0 = (vt == 0) ? 0 : (vt * 64 - 4);
        const int x1 = vt * 64 + 64;
#pragma unroll 2
        for (int xb = x0; xb < x1; xb += 4) {
            const int xk = xb + lhalf * 2;
            v2f a, b;
            a.x = T1[nrow * T1_STRIDE + xk];
            a.y = T1[nrow * T1_STRIDE + xk + 1];
            b.x = (xk     >= lokv && xk     <= hikv) ? 1.0f : 0.0f;
            b.y = (xk + 1 >= lokv && xk + 1 <= hikv) ? 1.0f : 0.0f;
            acc = __builtin_amdgcn_wmma_f32_16x16x4_f32(
                false, a, false, b, (short)0, acc, false, false);
        }
#pragma unroll
        for (int r = 0; r < 8; ++r) {
            const int n = nt * 16 + r + lhalf * 8;
            dsc[((size_t)((t * NQ + n) * NQ + vcol) << 6) + s] = acc[r];
        }
    }
}

// ---------------------------------------------------------------------------
// Kernel 3a: the box-sum identity breaks where reflection is active, i.e.
// queries with n in {0,1,63} or v in {0,1,63}. Recompute those directly.
// One block per query (64 threads = 64 shifts); interior blocks exit.
// ---------------------------------------------------------------------------
__global__ __launch_bounds__(64) void nls_fixup(const float* __restrict__ vid0,
                                                const float* __restrict__ vid1,
                                                float* __restrict__ dsc) {
    const int q = blockIdx.x;            // 0..12287
    const int t = q >> 12;
    const int n = (q >> 6) & 63;
    const int v = q & 63;
    const bool boundary = (n <= 1) || (n == 63) || (v <= 1) || (v == 63);
    if (!boundary) return;

    const int s   = threadIdx.x;         // 0..63
    const int s0p = (s >> 3) - 4;
    const int s1p = (s & 7) - 4;

    const float* b0 = vid0 + ((size_t)(t * CCH) << 16);
    const float* b1 = vid1 + ((size_t)(t * CCH) << 16);

    float acc = 0.0f;
    for (int i = 0; i < 7; ++i) {
        const int y0 = refl(4 * n + i - 3);
        const int y1 = refl(4 * n + i - 3 + s0p);
        for (int j = 0; j < 7; ++j) {
            const int x0 = refl(4 * v + j - 3);
            const int x1 = refl(4 * v + j - 3 + s1p);
            const float* p0 = b0 + y0 * HW + x0;
            const float* p1 = b1 + y1 * HW + x1;
#pragma unroll
            for (int c = 0; c < CCH; ++c)
                acc += p0[(size_t)c << 16] * p1[(size_t)c << 16];
        }
    }
    dsc[((size_t)q << 6) + s] = acc;
}

// ---------------------------------------------------------------------------
// Kernel 3b: top-7 per query. Self shift (index 36) gets +1e30 for ranking
// only; emitted value is the raw score. 7 stable argmax passes (lowest index
// wins ties, matching lax.top_k). Indices: (t, refl(4n+dh), refl(4v+dw)).
// ---------------------------------------------------------------------------
__global__ __launch_bounds__(256) void nls_topk(const float* __restrict__ dsc,
                                                float* __restrict__ dout,
                                                int* __restrict__ iout) {
    const int q = blockIdx.x * blockDim.x + threadIdx.x;
    if (q >= NQUERIES) return;
    const int t = q >> 12;
    const int n = (q >> 6) & 63;
    const int v = q & 63;
    const float* d = dsc + ((size_t)q << 6);

    unsigned long long sel = 0ull;
    for (int k = 0; k < KTOP; ++k) {
        float best = -3.0e38f;
        int   bidx = 0;
        for (int s = 0; s < NSHIFT; ++s) {
            if ((sel >> s) & 1ull) continue;
            const float a = d[s] + ((s == 36) ? 1e30f : 0.0f);
            if (a > best) { best = a; bidx = s; }
        }
        sel |= (1ull << bidx);
        dout[q * KTOP + k] = d[bidx];
        const int dh = (bidx >> 3) - 4;
        const int dw = (bidx & 7) - 4;
        int* ip = iout + (size_t)(q * KTOP + k) * 3;
        ip[0] = t;
        ip[1] = refl(4 * n + dh);
        ip[2] = refl(4 * v + dw);
    }
}

// ---------------------------------------------------------------------------
// Host-side launch. d_ws layout: [CC: 192*65536 f32][scores: 12288*64 f32].
// d_out layout: [dists: 12288*7 f32][inds: 12288*7*3 i32] (tuple concat).
// ---------------------------------------------------------------------------
extern "C" void kernel_launch(void* const* d_in, const int* in_sizes, int n_in,
                              void* d_out, int out_size, void* d_ws, size_t ws_size,
                              hipStream_t stream) {
    const float* vid0 = (const float*)d_in[0];
    const float* vid1 = (const float*)d_in[1];

    float* cc  = (float*)d_ws;
    float* dsc = cc + (size_t)TT * NSHIFT * HW * HW;   // + 12,582,912 floats

    float* dout = (float*)d_out;
    int*   iout = (int*)d_out + (size_t)NQUERIES * KTOP;

    nls_cc<<<dim3(HW, 8, TT), 256, 0, stream>>>(vid0, vid1, cc);
    nls_boxsum_wmma<<<TT * NSHIFT, 256, 64 * T1_STRIDE * sizeof(float), stream>>>(cc, dsc);
    nls_fixup<<<NQUERIES, 64, 0, stream>>>(vid0, vid1, dsc);
    nls_topk<<<(NQUERIES + 255) / 256, 256, 0, stream>>>(dsc, dout, iout);
}